// FinalLayer_19696720019875
// MI455X (gfx1250) — compile-verified
//
#include <hip/hip_runtime.h>

typedef __attribute__((ext_vector_type(2))) float v2f;
typedef __attribute__((ext_vector_type(8))) float v8f;

#define NB 4
#define NG 1024
#define NT 1024
#define NK 5
#define NC 8
#define GSPLIT 4          // waves per block, each takes NG/GSPLIT grid points
#define LOG2E 1.4426950408889634f
#define EPSF  1e-6f

// Workspace layout (floats): xt[b][c][g] | yt[b][c][t] | ht[b][c][k][g]
#define XT_OFF 0
#define XT_SZ  (NB * NC * NG)                 // 32768
#define YT_OFF (XT_OFF + XT_SZ)
#define YT_SZ  (NB * NC * NT)                 // 32768
#define HT_OFF (YT_OFF + YT_SZ)
#define HT_SZ  (NB * NC * NK * NG)            // 163840
#define WS_FLOATS (HT_OFF + HT_SZ)            // 229376 floats = 896 KB

// ---------------------------------------------------------------------------
// Stage 0: re-layout operands so the g axis is innermost (unit stride).
// Reads are strided, writes coalesced; total traffic ~1.5 MB -> negligible.
// ---------------------------------------------------------------------------
__global__ __launch_bounds__(256) void stage_transpose(
    const float* __restrict__ x_grid,    // (B,G,C)
    const float* __restrict__ h_grid,    // (B,G,K,C)
    const float* __restrict__ target_x,  // (B,T,C)
    float* __restrict__ ws)
{
    const int i = blockIdx.x * blockDim.x + threadIdx.x;

    if (i < XT_SZ) {                      // xt[b][c][g] = x[b][g][c]
        const int g = i & (NG - 1);
        const int c = (i / NG) & (NC - 1);
        const int b = i / (NG * NC);
        ws[XT_OFF + i] = x_grid[((size_t)b * NG + g) * NC + c];
    }
    if (i < YT_SZ) {                      // yt[b][c][t] = y[b][t][c]
        const int t = i & (NT - 1);
        const int c = (i / NT) & (NC - 1);
        const int b = i / (NT * NC);
        ws[YT_OFF + i] = target_x[((size_t)b * NT + t) * NC + c];
    }
    if (i < HT_SZ) {                      // ht[b][c][k][g] = h[b][g][k][c]
        const int g = i & (NG - 1);
        const int k = (i / NG) % NK;
        const int c = (i / (NG * NK)) & (NC - 1);
        const int b = i / (NG * NK * NC);
        ws[HT_OFF + i] = h_grid[(((size_t)b * NG + g) * NK + k) * NC + c];
    }
}

// ---------------------------------------------------------------------------
// Main kernel (staged layout). Block = 4 wave32s handling (b, c, 16-t tile);
// wave w covers g in [w*256, w*256+256). Per 4-g step: one b64 x-load, shared
// squared diffs, then per basis k: one b64 h-load, 2x (v_mul + v_exp_f32) and
// one v_wmma_f32_16x16x4_f32 (B = h broadcast over N columns).
// ---------------------------------------------------------------------------
__global__ __launch_bounds__(32 * GSPLIT) void gabor_conv_wmma_staged(
    const float* __restrict__ ws,
    const float* __restrict__ sigma,     // (K,C)
    float* __restrict__ out)             // (B,T,K,C)
{
    __shared__ float red[GSPLIT][NK][16];

    const int lane = threadIdx.x & 31;
    const int wave = threadIdx.x >> 5;
    const int t0   = blockIdx.x * 16;
    const int c    = blockIdx.y;
    const int b    = blockIdx.z;

    const int m   = lane & 15;            // A row (target index) / D column
    const int kh  = lane >> 4;            // half-wave -> K slots {0,1} vs {2,3}
    const int kg0 = 2 * kh;

    float ck2[NK];
#pragma unroll
    for (int k = 0; k < NK; ++k) {
        float s = __expf(sigma[k * NC + c]) + EPSF;
        ck2[k] = (-0.5f * LOG2E) / (s * s);
    }

    const float yv = ws[YT_OFF + ((size_t)b * NC + c) * NT + t0 + m];

    const int g0 = wave * (NG / GSPLIT);
    const float* xp = ws + XT_OFF + ((size_t)b * NC + c) * NG + g0;
    const float* hp = ws + HT_OFF + (((size_t)b * NC + c) * NK) * NG + g0; // + k*NG

    v8f acc0 = {}, acc1 = {}, acc2 = {}, acc3 = {}, acc4 = {};

    for (int g = 0; g < NG / GSPLIT; g += 4) {
        const v2f xv = *(const v2f*)(xp + g + kg0);      // x[g+kg0], x[g+kg0+1]
        const float d0 = xv.x - yv;
        const float d1 = xv.y - yv;
        const float s0 = d0 * d0;
        const float s1 = d1 * d1;

#define DO_K(KK, ACC)                                                          \
        {                                                                      \
            v2f a;                                                             \
            const v2f bb = *(const v2f*)(hp + (KK) * NG + g + kg0);            \
            a.x = __builtin_amdgcn_exp2f(s0 * ck2[KK]);                        \
            a.y = __builtin_amdgcn_exp2f(s1 * ck2[KK]);                        \
            ACC = __builtin_amdgcn_wmma_f32_16x16x4_f32(                       \
                false, a, false, bb, (short)0, ACC, false, false);             \
        }
        DO_K(0, acc0)
        DO_K(1, acc1)
        DO_K(2, acc2)
        DO_K(3, acc3)
        DO_K(4, acc4)
#undef DO_K
    }

    if (m == 0) {                         // N==0 lanes hold rows kh*8..kh*8+7
        const int mb = kh * 8;
#pragma unroll
        for (int r = 0; r < 8; ++r) {
            red[wave][0][mb + r] = acc0[r];
            red[wave][1][mb + r] = acc1[r];
            red[wave][2][mb + r] = acc2[r];
            red[wave][3][mb + r] = acc3[r];
            red[wave][4][mb + r] = acc4[r];
        }
    }
    __syncthreads();

    const int idx = threadIdx.x;
    if (idx < NK * 16) {
        const int k = idx >> 4;
        const int r = idx & 15;
        const float sum = red[0][k][r] + red[1][k][r] + red[2][k][r] + red[3][k][r];
        out[(((size_t)b * NT + (t0 + r)) * NK + k) * NC + c] = sum;
    }
}

// ---------------------------------------------------------------------------
// Fallback (no scratch): identical math directly on the (…,c)-innermost layout.
// ---------------------------------------------------------------------------
__global__ __launch_bounds__(32 * GSPLIT) void gabor_conv_wmma_direct(
    const float* __restrict__ x_grid,
    const float* __restrict__ h_grid,
    const float* __restrict__ target_x,
    const float* __restrict__ sigma,
    float* __restrict__ out)
{
    __shared__ float red[GSPLIT][NK][16];

    const int lane = threadIdx.x & 31;
    const int wave = threadIdx.x >> 5;
    const int t0   = blockIdx.x * 16;
    const int c    = blockIdx.y;
    const int b    = blockIdx.z;

    const int m   = lane & 15;
    const int kh  = lane >> 4;
    const int kg0 = 2 * kh;

    float ck2[NK];
#pragma unroll
    for (int k = 0; k < NK; ++k) {
        float s = __expf(sigma[k * NC + c]) + EPSF;
        ck2[k] = (-0.5f * LOG2E) / (s * s);
    }

    const float yv = target_x[((size_t)b * NT + (t0 + m)) * NC + c];

    const int g0 = wave * (NG / GSPLIT);
    const float* xp = x_grid + ((size_t)b * NG + g0) * NC + c;
    const float* hp = h_grid + (((size_t)b * NG + g0) * NK) * NC + c;

    v8f acc0 = {}, acc1 = {}, acc2 = {}, acc3 = {}, acc4 = {};

    for (int g = 0; g < NG / GSPLIT; g += 4) {
        const float xv0 = xp[(g + kg0    ) * NC];
        const float xv1 = xp[(g + kg0 + 1) * NC];
        const float d0 = xv0 - yv;
        const float d1 = xv1 - yv;
        const float s0 = d0 * d0;
        const float s1 = d1 * d1;

        const float* hbase = hp + (size_t)(g + kg0) * NK * NC;

#define DO_K(KK, ACC)                                                          \
        {                                                                      \
            v2f a, bb;                                                         \
            a.x = __builtin_amdgcn_exp2f(s0 * ck2[KK]);                        \
            a.y = __builtin_amdgcn_exp2f(s1 * ck2[KK]);                        \
            bb.x = hbase[(KK) * NC];                                           \
            bb.y = hbase[NK * NC + (KK) * NC];                                 \
            ACC = __builtin_amdgcn_wmma_f32_16x16x4_f32(                       \
                false, a, false, bb, (short)0, ACC, false, false);             \
        }
        DO_K(0, acc0)
        DO_K(1, acc1)
        DO_K(2, acc2)
        DO_K(3, acc3)
        DO_K(4, acc4)
#undef DO_K
    }

    if (m == 0) {
        const int mb = kh * 8;
#pragma unroll
        for (int r = 0; r < 8; ++r) {
            red[wave][0][mb + r] = acc0[r];
            red[wave][1][mb + r] = acc1[r];
            red[wave][2][mb + r] = acc2[r];
            red[wave][3][mb + r] = acc3[r];
            red[wave][4][mb + r] = acc4[r];
        }
    }
    __syncthreads();

    const int idx = threadIdx.x;
    if (idx < NK * 16) {
        const int k = idx >> 4;
        const int r = idx & 15;
        const float sum = red[0][k][r] + red[1][k][r] + red[2][k][r] + red[3][k][r];
        out[(((size_t)b * NT + (t0 + r)) * NK + k) * NC + c] = sum;
    }
}

extern "C" void kernel_launch(void* const* d_in, const int* in_sizes, int n_in,
                              void* d_out, int out_size, void* d_ws, size_t ws_size,
                              hipStream_t stream) {
    const float* x_grid   = (const float*)d_in[0];
    const float* h_grid   = (const float*)d_in[1];
    const float* target_x = (const float*)d_in[2];
    const float* sigma    = (const float*)d_in[3];
    float* out = (float*)d_out;

    dim3 grid(NT / 16, NC, NB);   // 64 target tiles x 8 channels x 4 batches
    dim3 block(32 * GSPLIT);      // 4 wave32s per block

    if (ws_size >= (size_t)WS_FLOATS * sizeof(float)) {
        float* ws = (float*)d_ws;
        stage_transpose<<<(HT_SZ + 255) / 256, 256, 0, stream>>>(
            x_grid, h_grid, target_x, ws);
        gabor_conv_wmma_staged<<<grid, block, 0, stream>>>(ws, sigma, out);
    } else {
        gabor_conv_wmma_direct<<<grid, block, 0, stream>>>(
            x_grid, h_grid, target_x, sigma, out);
    }
}